// End2EndQNetwork_24068996726956
// MI455X (gfx1250) — compile-verified
//
#include <hip/hip_runtime.h>
#include <hip/hip_bf16.h>
#include <math.h>

// ---------------------------------------------------------------------------
// Problem constants (from reference)
// ---------------------------------------------------------------------------
#define E_DIM   128
#define T_NODES 120          // 2*GT + 2*OBJ + ENTS + OPS
#define LEAVES  16
#define GT      32
#define TH      32
#define OBJ     8
#define ENTS    32
#define OPS     8
#define CONC_ROWS 113        // (GT+TH+OBJ+ENTS+1+OPS)
#define CONC    (CONC_ROWS * E_DIM)   // 14464
#define Q1_OUT  (CONC / 2)            // 7232
#define Q2_OUT  (CONC / 4)            // 3616

typedef float v2f __attribute__((ext_vector_type(2)));
typedef float v4f __attribute__((ext_vector_type(4)));
typedef float v8f __attribute__((ext_vector_type(8)));

// ---------------------------------------------------------------------------
// Shared per-function MLP core:  y = W2[fid] @ relu(W1[fid] @ x + b1) + b2
//   W1: (512,256)  W2: (128,512)    x: 256 in LDS, h: 512 in LDS, y: 128 LDS
// Executed by a 128-thread (4-wave) block.  Uses V_WMMA_F32_16X16X4_F32 as a
// 16-row GEMV tile: A = 16x4 weight tile; B = x[k..k+3] broadcast into all 16
// columns; D column 0 holds the partial dot products (f32 exact).
//
// f32 16x4 A layout (ISA 7.12.2): lanes 0-15 hold rows M=lane, K={0,1};
// lanes 16-31 hold rows M=lane-16, K={2,3}.  B mirrors with N=lane%16 and we
// replicate x across N, so each lane only needs the K values of its half.
// D layout: VGPR r: lanes 0-15 -> M=r, N=lane; lanes 16-31 -> M=r+8.
// ---------------------------------------------------------------------------
__device__ __forceinline__ void per_fn_mlp(const float* __restrict__ w1,
                                           const float* __restrict__ b1,
                                           const float* __restrict__ w2,
                                           const float* __restrict__ b2,
                                           int fid,
                                           const float* __restrict__ xs,  // [256] LDS
                                           float* __restrict__ hs,        // [512] LDS
                                           float* __restrict__ ys)        // [128] LDS
{
    const int tid   = threadIdx.x;
    const int wave  = tid >> 5;
    const int lane  = tid & 31;
    const int lhalf = lane >> 4;       // 0: K sub 0,1   1: K sub 2,3
    const int mlane = lane & 15;
    const int kh    = lhalf * 2;

    // ---- Layer 1: h = relu(W1 @ x + b1), 512 rows = 4 waves * 8 tiles * 16
    const float* W1 = w1 + (size_t)fid * 512 * 256;
    const float* B1 = b1 + (size_t)fid * 512;
    for (int tile = 0; tile < 8; ++tile) {
        const int m0 = wave * 128 + tile * 16;
        const float* wrow = W1 + (size_t)(m0 + mlane) * 256;
        v8f acc = {0.f, 0.f, 0.f, 0.f, 0.f, 0.f, 0.f, 0.f};
        #pragma unroll 4
        for (int k0 = 0; k0 < 256; k0 += 4) {
            v2f a, b;
            a.x = wrow[k0 + kh];
            a.y = wrow[k0 + kh + 1];
            b.x = xs[k0 + kh];
            b.y = xs[k0 + kh + 1];
            acc = __builtin_amdgcn_wmma_f32_16x16x4_f32(
                      false, a, false, b, (short)0, acc, false, false);
        }
        if (mlane == 0) {                  // lane 0 -> rows m0..m0+7, lane 16 -> m0+8..m0+15
            const int rb = m0 + lhalf * 8;
            #pragma unroll
            for (int r = 0; r < 8; ++r)
                hs[rb + r] = fmaxf(acc[r] + B1[rb + r], 0.f);
        }
    }
    __syncthreads();

    // ---- Layer 2: y = W2 @ h + b2, 128 rows = 4 waves * 2 tiles * 16
    const float* W2 = w2 + (size_t)fid * 128 * 512;
    const float* B2 = b2 + (size_t)fid * 128;
    for (int tile = 0; tile < 2; ++tile) {
        const int m0 = wave * 32 + tile * 16;
        const float* wrow = W2 + (size_t)(m0 + mlane) * 512;
        v8f acc = {0.f, 0.f, 0.f, 0.f, 0.f, 0.f, 0.f, 0.f};
        #pragma unroll 4
        for (int k0 = 0; k0 < 512; k0 += 4) {
            v2f a, b;
            a.x = wrow[k0 + kh];
            a.y = wrow[k0 + kh + 1];
            b.x = hs[k0 + kh];
            b.y = hs[k0 + kh + 1];
            acc = __builtin_amdgcn_wmma_f32_16x16x4_f32(
                      false, a, false, b, (short)0, acc, false, false);
        }
        if (mlane == 0) {
            const int rb = m0 + lhalf * 8;
            #pragma unroll
            for (int r = 0; r < 8; ++r)
                ys[rb + r] = acc[r] + B2[rb + r];
        }
    }
    __syncthreads();
}

// ---------------------------------------------------------------------------
// Kernel 1: gather leaf embeddings.  h0[g] = entity_emb[leaf_idx[g]]
// grid = T*LEAVES blocks of 32 threads; 128-bit copy of 128 floats.
// ---------------------------------------------------------------------------
__global__ void gather_leaves(const int* __restrict__ leaf_idx,
                              const float* __restrict__ emb,
                              float* __restrict__ h0)
{
    const int g = blockIdx.x;
    const int idx = leaf_idx[g];
    const v4f* src = (const v4f*)(emb + (size_t)idx * E_DIM);
    v4f* dst = (v4f*)(h0 + (size_t)g * E_DIM);
    dst[threadIdx.x] = src[threadIdx.x];
}

// ---------------------------------------------------------------------------
// Kernel 2: one tree level.  in: (T, 2n, 128) -> out: (T, n, 128)
// x[t,j] = concat(in[t,2j], in[t,2j+1])  (contiguous 256 floats)
// grid = T*n blocks of 128 threads.
// ---------------------------------------------------------------------------
__global__ void tree_level(const float* __restrict__ in,
                           float* __restrict__ out,
                           const int* __restrict__ nf_fids,
                           int fid_off, int n,
                           const float* __restrict__ w1, const float* __restrict__ b1,
                           const float* __restrict__ w2, const float* __restrict__ b2)
{
    __shared__ float xs[256];
    __shared__ float hs[512];
    __shared__ float ys[128];

    const int g = blockIdx.x;
    const int t = g / n;
    const int j = g % n;
    const int fid = nf_fids[t * 15 + fid_off + j];
    const float* xsrc = in + ((size_t)t * 2 * n + 2 * j) * E_DIM;

    for (int i = threadIdx.x; i < 256; i += 128) xs[i] = xsrc[i];
    __syncthreads();

    per_fn_mlp(w1, b1, w2, b2, fid, xs, hs, ys);

    out[((size_t)t * n + j) * E_DIM + threadIdx.x] = ys[threadIdx.x];
}

// ---------------------------------------------------------------------------
// Kernel 3: leaf MLPs with tanh, written directly into the concat x vector.
// blocks 0..31: gt node j  (roots rows 2j,2j+1)      -> xv rows j
// blocks 32..39: obj node j (roots rows 64+2j,65+2j) -> xv rows 64+j
// ---------------------------------------------------------------------------
__global__ void leaf_mlp(const float* __restrict__ roots,
                         const int* __restrict__ lf_fids,
                         const float* __restrict__ w1, const float* __restrict__ b1,
                         const float* __restrict__ w2, const float* __restrict__ b2,
                         float* __restrict__ xv)
{
    __shared__ float xs[256];
    __shared__ float hs[512];
    __shared__ float ys[128];

    const int nb = blockIdx.x;
    int fid;
    const float* xsrc;
    float* dst;
    if (nb < GT) {
        fid  = lf_fids[nb];
        xsrc = roots + (size_t)(2 * nb) * E_DIM;
        dst  = xv + (size_t)nb * E_DIM;
    } else {
        const int j = nb - GT;
        fid  = lf_fids[GT + j];
        xsrc = roots + (size_t)(2 * GT + 2 * j) * E_DIM;
        dst  = xv + (size_t)(2 * GT + j) * E_DIM;     // rows 64..71
    }

    for (int i = threadIdx.x; i < 256; i += 128) xs[i] = xsrc[i];
    __syncthreads();

    per_fn_mlp(w1, b1, w2, b2, fid, xs, hs, ys);

    dst[threadIdx.x] = tanhf(ys[threadIdx.x]);
}

// ---------------------------------------------------------------------------
// Kernel 4: assemble the remaining rows of x:
//   rows 32..63  = theorem_emb[th_idx]
//   rows 72..103 = roots[80..111]   (ents)
//   row  104     = theorem_emb[act_th_idx[0]]
//   rows 105..112= roots[112..119]  (ops)
// grid = 73 blocks of 32 threads (128-bit copy).
// ---------------------------------------------------------------------------
__global__ void assemble_x(const float* __restrict__ roots,
                           const float* __restrict__ th_emb,
                           const int* __restrict__ th_idx,
                           const int* __restrict__ act_idx,
                           float* __restrict__ xv)
{
    const int b = blockIdx.x;
    const float* src;
    float* dst;
    if (b < 32) {                       // th
        src = th_emb + (size_t)th_idx[b] * E_DIM;
        dst = xv + (size_t)(32 + b) * E_DIM;
    } else if (b < 64) {                // ents
        const int j = b - 32;
        src = roots + (size_t)(80 + j) * E_DIM;
        dst = xv + (size_t)(72 + j) * E_DIM;
    } else if (b == 64) {               // act theorem
        src = th_emb + (size_t)act_idx[0] * E_DIM;
        dst = xv + (size_t)104 * E_DIM;
    } else {                            // ops
        const int j = b - 65;
        src = roots + (size_t)(112 + j) * E_DIM;
        dst = xv + (size_t)(105 + j) * E_DIM;
    }
    ((v4f*)dst)[threadIdx.x] = ((const v4f*)src)[threadIdx.x];
}

// ---------------------------------------------------------------------------
// Kernel 5: row-parallel GEMV  y[i] = act(W[i,:] . x + b[i]).
// One wave per row, 128-bit loads, wave32 shuffle reduction.
// NT=true streams W non-temporally (q_w1 is 418 MB > 192 MB L2: single-use,
// keep it out of L2 so q_w2 + tree weights stay resident across replays).
// ---------------------------------------------------------------------------
template <bool NT, bool RELU>
__global__ void matvec_rows(const float* __restrict__ W,
                            const float* __restrict__ bias,
                            const float* __restrict__ x,
                            float* __restrict__ y,
                            int rows, int cols)
{
    const int row  = blockIdx.x * (blockDim.x >> 5) + (threadIdx.x >> 5);
    const int lane = threadIdx.x & 31;
    if (row >= rows) return;

    const float* wr = W + (size_t)row * cols;
    float s = 0.f;

    const int limit = cols & ~127;          // multiple of 32 lanes * 4
    for (int j = lane * 4; j < limit; j += 128) {
        v4f wv;
        if (NT) wv = __builtin_nontemporal_load((const v4f*)(wr + j));
        else    wv = *(const v4f*)(wr + j);
        const v4f xv = *(const v4f*)(x + j);
        s += wv.x * xv.x + wv.y * xv.y + wv.z * xv.z + wv.w * xv.w;
    }
    for (int k = limit + lane; k < cols; k += 32)
        s += wr[k] * x[k];

    #pragma unroll
    for (int off = 16; off > 0; off >>= 1)
        s += __shfl_down(s, off, 32);

    if (lane == 0) {
        float v = s + bias[row];
        y[row] = RELU ? fmaxf(v, 0.f) : v;
    }
}

// ---------------------------------------------------------------------------
// Launch
// ---------------------------------------------------------------------------
extern "C" void kernel_launch(void* const* d_in, const int* in_sizes, int n_in,
                              void* d_out, int out_size, void* d_ws, size_t ws_size,
                              hipStream_t stream)
{
    (void)in_sizes; (void)n_in; (void)out_size; (void)ws_size;

    const int*   leaf_idx    = (const int*)  d_in[0];
    const int*   nf_fids     = (const int*)  d_in[1];
    const int*   lf_fids     = (const int*)  d_in[2];
    const int*   th_idx      = (const int*)  d_in[3];
    const int*   act_th_idx  = (const int*)  d_in[4];
    const float* entity_emb  = (const float*)d_in[5];
    const float* theorem_emb = (const float*)d_in[6];
    const float* nf_w1       = (const float*)d_in[7];
    const float* nf_b1       = (const float*)d_in[8];
    const float* nf_w2       = (const float*)d_in[9];
    const float* nf_b2       = (const float*)d_in[10];
    const float* lf_w1       = (const float*)d_in[11];
    const float* lf_b1       = (const float*)d_in[12];
    const float* lf_w2       = (const float*)d_in[13];
    const float* lf_b2       = (const float*)d_in[14];
    const float* q_w1        = (const float*)d_in[15];
    const float* q_b1        = (const float*)d_in[16];
    const float* q_w2        = (const float*)d_in[17];
    const float* q_b2        = (const float*)d_in[18];
    const float* q_w3        = (const float*)d_in[19];
    const float* q_b3        = (const float*)d_in[20];
    float* out = (float*)d_out;

    // Workspace layout (floats), ~2 MB total
    float* ws    = (float*)d_ws;
    float* bufA  = ws;                              // 120*16*128 = 245760
    float* bufB  = bufA + (size_t)T_NODES * 16 * E_DIM;   // 120*8*128
    float* bufC  = bufB + (size_t)T_NODES *  8 * E_DIM;   // 120*4*128
    float* bufD  = bufC + (size_t)T_NODES *  4 * E_DIM;   // 120*2*128
    float* roots = bufD + (size_t)T_NODES *  2 * E_DIM;   // 120*128
    float* xv    = roots + (size_t)T_NODES * E_DIM;       // 113*128
    float* qh1   = xv + CONC;                             // 7232
    float* qh2   = qh1 + Q1_OUT;                          // 3616

    // 1) gather leaves
    gather_leaves<<<T_NODES * LEAVES, 32, 0, stream>>>(leaf_idx, entity_emb, bufA);

    // 2) tree reduction: 16 -> 8 -> 4 -> 2 -> 1
    tree_level<<<T_NODES * 8, 128, 0, stream>>>(bufA, bufB, nf_fids,  0, 8,
                                                nf_w1, nf_b1, nf_w2, nf_b2);
    tree_level<<<T_NODES * 4, 128, 0, stream>>>(bufB, bufC, nf_fids,  8, 4,
                                                nf_w1, nf_b1, nf_w2, nf_b2);
    tree_level<<<T_NODES * 2, 128, 0, stream>>>(bufC, bufD, nf_fids, 12, 2,
                                                nf_w1, nf_b1, nf_w2, nf_b2);
    tree_level<<<T_NODES * 1, 128, 0, stream>>>(bufD, roots, nf_fids, 14, 1,
                                                nf_w1, nf_b1, nf_w2, nf_b2);

    // 3) leaf MLPs (gt + obj, tanh) straight into x
    leaf_mlp<<<GT + OBJ, 128, 0, stream>>>(roots, lf_fids,
                                           lf_w1, lf_b1, lf_w2, lf_b2, xv);

    // 4) remaining rows of x
    assemble_x<<<73, 32, 0, stream>>>(roots, theorem_emb, th_idx, act_th_idx, xv);

    // 5) Q head: q1 streams 418 MB non-temporally (HBM-bound), q2 L2-friendly
    matvec_rows<true,  true ><<<Q1_OUT / 8, 256, 0, stream>>>(q_w1, q_b1, xv,  qh1,
                                                              Q1_OUT, CONC);
    matvec_rows<false, true ><<<Q2_OUT / 8, 256, 0, stream>>>(q_w2, q_b2, qh1, qh2,
                                                              Q2_OUT, Q1_OUT);
    matvec_rows<false, false><<<1, 32, 0, stream>>>(q_w3, q_b3, qh2, out,
                                                    1, Q2_OUT);
}